// AdaptiveTokenMerger_63153199120660
// MI455X (gfx1250) — compile-verified
//
#include <hip/hip_runtime.h>

typedef float v2f __attribute__((ext_vector_type(2)));
typedef float v8f __attribute__((ext_vector_type(8)));

// Problem constants: B=4, D=768, H=224, W=224, T=14, g=16, N=196, L=256, heads=12, hd=64
#define HW_   50176
#define CHW_  38535168LL   // 768*50176
#define ROW3  150528LL     // 3*50176  (kv row stride over l)

// kv[l,c] element offset within x for the "ATM" reinterpretation:
//   koff(c) = (c>>8)*HW + ((c>>4)&15)*224 + (c&15)
__device__ __forceinline__ long long atm_koff(int c) {
  return (long long)((c >> 8) * HW_ + (((c >> 4) & 15) * 224) + (c & 15));
}
// per-(b,n) base into x:  z = b*196 + n, n = trow*14 + tcol
__device__ __forceinline__ long long atm_base(int z) {
  int b = z / 196, n = z % 196;
  int tr = n / 14, tc = n % 14;
  return (long long)b * CHW_ + (long long)(tr * 3584 + tc * 16);
}

// gfx1250 async global->LDS copy (per-lane, tracked by ASYNCcnt).
// vdst = LDS byte address (low 32 bits of generic shared pointer), vaddr = 64-bit global VA.
__device__ __forceinline__ void async_g2l_b32(const float* gptr, float* lptr) {
  unsigned lds_off = (unsigned)(unsigned long long)lptr;   // flat->LDS keeps addr[31:0]
  unsigned long long ga = (unsigned long long)gptr;
  asm volatile("global_load_async_to_lds_b32 %0, %1, off"
               :: "v"(lds_off), "v"(ga) : "memory");
}
__device__ __forceinline__ void wait_async0() {
  asm volatile("s_wait_asynccnt 0x0" ::: "memory");
}

// ---------------------------------------------------------------------------
// Generic f32 WMMA GEMM:
//   C[m,n] = act( alpha * sum_k A(m,k)*B(n,k) + bias[n] )
// elem(A,m,k) = A[batchBaseA + rowOffA(m) + kOffA(k)]
//   rowOffA(m) = aRowAtm ? atm_koff(m) : m*aRs
//   kOffA(k)   = aKAtm   ? atm_koff(k) : k*aKs
//   batchBaseA = (aMode==1) ? atm_base(z) : z*aBatch
// (same scheme for B).  C addr = z*cBatch + m*cMs + n*cNs.
// K must be a multiple of 32. Block = 128 threads (4 waves), tile 64x64.
// Slab staging uses GLOBAL_LOAD_ASYNC_TO_LDS_B32 (ASYNCcnt), drained before
// the workgroup barrier publishes the LDS tile.
// ---------------------------------------------------------------------------
#define KC   32
#define LDSK 34

__global__ __launch_bounds__(128)
void gemm_wmma_f32(
    const float* __restrict__ A, long long aBatch, int aMode,
    long long aRs, long long aKs, int aRowAtm, int aKAtm,
    const float* __restrict__ B, long long bBatch, int bMode,
    long long bRs, long long bKs, int bRowAtm, int bKAtm,
    float* __restrict__ C, long long cBatch, long long cMs, long long cNs,
    const float* __restrict__ bias, long long biasBatch,
    int M, int N, int K, float alpha, int act)
{
  __shared__ float lds_a[64 * LDSK];
  __shared__ float lds_b[64 * LDSK];

  const int tid  = threadIdx.x;
  const int lane = tid & 31;          // wave32
  const int wave = tid >> 5;
  const int z  = blockIdx.z;
  const int bm = blockIdx.x, bn = blockIdx.y;

  const long long baseA = (aMode == 1) ? atm_base(z) : (long long)z * aBatch;
  const long long baseB = (bMode == 1) ? atm_base(z) : (long long)z * bBatch;
  const long long baseC = (long long)z * cBatch;

  const int wm = (wave >> 1) * 32;    // wave's 32x32 sub-tile
  const int wn = (wave & 1) * 32;

  v8f acc[2][2];
#pragma unroll
  for (int i = 0; i < 2; ++i)
#pragma unroll
    for (int j = 0; j < 2; ++j)
      acc[i][j] = (v8f){0.f,0.f,0.f,0.f,0.f,0.f,0.f,0.f};

  for (int ks = 0; ks < K; ks += KC) {
    // ---- stage A slab: 64 rows x 32 k (async global -> LDS) ----
    for (int idx = tid; idx < 64 * KC; idx += 128) {
      int r = idx >> 5, k = idx & (KC - 1);
      int gm = bm * 64 + r; if (gm >= M) gm = M - 1;
      int gk = ks + k;
      long long off = baseA
        + (aRowAtm ? atm_koff(gm) : (long long)gm * aRs)
        + (aKAtm   ? atm_koff(gk) : (long long)gk * aKs);
      async_g2l_b32(A + off, &lds_a[r * LDSK + k]);
    }
    // ---- stage B slab: 64 rows x 32 k (async global -> LDS) ----
    for (int idx = tid; idx < 64 * KC; idx += 128) {
      int r = idx >> 5, k = idx & (KC - 1);
      int gn = bn * 64 + r; if (gn >= N) gn = N - 1;
      int gk = ks + k;
      long long off = baseB
        + (bRowAtm ? atm_koff(gn) : (long long)gn * bRs)
        + (bKAtm   ? atm_koff(gk) : (long long)gk * bKs);
      async_g2l_b32(B + off, &lds_b[r * LDSK + k]);
    }
    wait_async0();       // drain this wave's async copies
    __syncthreads();     // publish LDS tile to all waves

#pragma unroll
    for (int k0 = 0; k0 < KC; k0 += 4) {
      // 16x4 f32 A layout: lanes 0-15 hold K = k0,k0+1 ; lanes 16-31 hold K = k0+2,k0+3
      const int khi = k0 + ((lane >> 4) << 1);
      const int mrw = lane & 15;
      const float* pa0 = &lds_a[(wm      + mrw) * LDSK + khi];
      const float* pa1 = &lds_a[(wm + 16 + mrw) * LDSK + khi];
      const float* pb0 = &lds_b[(wn      + mrw) * LDSK + khi];
      const float* pb1 = &lds_b[(wn + 16 + mrw) * LDSK + khi];
      v2f a0; a0.x = pa0[0]; a0.y = pa0[1];
      v2f a1; a1.x = pa1[0]; a1.y = pa1[1];
      v2f b0; b0.x = pb0[0]; b0.y = pb0[1];
      v2f b1; b1.x = pb1[0]; b1.y = pb1[1];

      acc[0][0] = __builtin_amdgcn_wmma_f32_16x16x4_f32(false, a0, false, b0, (short)0, acc[0][0], false, false);
      acc[0][1] = __builtin_amdgcn_wmma_f32_16x16x4_f32(false, a0, false, b1, (short)0, acc[0][1], false, false);
      acc[1][0] = __builtin_amdgcn_wmma_f32_16x16x4_f32(false, a1, false, b0, (short)0, acc[1][0], false, false);
      acc[1][1] = __builtin_amdgcn_wmma_f32_16x16x4_f32(false, a1, false, b1, (short)0, acc[1][1], false, false);
    }
    __syncthreads();
  }

  // ---- epilogue: C 16x16 f32 layout: VGPR r -> M = r (lanes 0-15) / r+8 (lanes 16-31), N = lane&15
  const int mhalf = (lane >> 4) << 3;
#pragma unroll
  for (int i = 0; i < 2; ++i) {
#pragma unroll
    for (int j = 0; j < 2; ++j) {
      int n = bn * 64 + wn + j * 16 + (lane & 15);
      if (n >= N) continue;
      float bv = bias ? bias[(long long)z * biasBatch + n] : 0.f;
#pragma unroll
      for (int r = 0; r < 8; ++r) {
        int m = bm * 64 + wm + i * 16 + mhalf + r;
        if (m >= M) continue;
        float v = acc[i][j][r] * alpha + bv;
        if (act == 1) v = v / (1.f + __expf(-v));   // SiLU
        C[baseC + (long long)m * cMs + (long long)n * cNs] = v;
      }
    }
  }
}

// ---------------------------------------------------------------------------
// q[b, n, c] = mean over 16x16 window of x[b, c, :, :]
// block = (b, c, trow) ; 224 threads, one per column
// ---------------------------------------------------------------------------
__global__ void pool_q(const float* __restrict__ x, float* __restrict__ q)
{
  int id = blockIdx.x;
  int trow = id % 14; id /= 14;
  int c = id % 768;
  int b = id / 768;
  int w = threadIdx.x;  // 0..223

  const float* px = x + ((long long)b * 768 + c) * HW_ + (long long)(trow * 16) * 224 + w;
  float s = 0.f;
#pragma unroll
  for (int gy = 0; gy < 16; ++gy) s += px[gy * 224];

  __shared__ float col[224];
  col[w] = s;
  __syncthreads();
  if (w < 14) {
    float t = 0.f;
#pragma unroll
    for (int i = 0; i < 16; ++i) t += col[w * 16 + i];
    q[((long long)b * 196 + trow * 14 + w) * 768 + c] = t * (1.f / 256.f);
  }
}

// ---------------------------------------------------------------------------
// softmax over l (256) for scores laid out as [bn][l][16] (h in last dim, 12 used)
// one thread per (bn, h) pair; 784*12 = 9408 pairs
// ---------------------------------------------------------------------------
__global__ void softmax_attn(float* __restrict__ s)
{
  int idx = blockIdx.x * blockDim.x + threadIdx.x;
  if (idx >= 784 * 12) return;
  int bn = idx / 12, h = idx % 12;
  float* p = s + (long long)bn * 4096 + h;
  float mx = -3.4e38f;
  for (int l = 0; l < 256; ++l) { float v = p[l * 16]; mx = v > mx ? v : mx; }
  float sum = 0.f;
  for (int l = 0; l < 256; ++l) { float e = __expf(p[l * 16] - mx); p[l * 16] = e; sum += e; }
  float inv = 1.f / sum;
  for (int l = 0; l < 256; ++l) p[l * 16] *= inv;
}

// ---------------------------------------------------------------------------
extern "C" void kernel_launch(void* const* d_in, const int* in_sizes, int n_in,
                              void* d_out, int out_size, void* d_ws, size_t ws_size,
                              hipStream_t stream) {
  (void)in_sizes; (void)n_in; (void)out_size; (void)ws_size;
  const float* x     = (const float*)d_in[0];
  const float* w_in  = (const float*)d_in[1];
  const float* b_in  = (const float*)d_in[2];
  const float* w_out = (const float*)d_in[3];
  const float* b_out = (const float*)d_in[4];
  const float* w_fc1 = (const float*)d_in[5];
  const float* b_fc1 = (const float*)d_in[6];
  const float* w_fc2 = (const float*)d_in[7];
  const float* b_fc2 = (const float*)d_in[8];
  float* y  = (float*)d_out;
  float* ws = (float*)d_ws;

  // workspace layout (floats); ctx/out/h1 reuse dead buffers
  float* q     = ws;                    // 784 x 768
  float* qp    = q     + 602112;        // 784 x 768
  float* qk    = qp    + 602112;        // [bn][12][768]  (scale folded in)
  float* attn  = qk    + 7225344;       // [bn][256][16]
  float* kvbar = attn  + 3211264;       // [bn][12][768]
  float* ctx   = q;                     // reuse (q dead after step 2)
  float* outp  = qp;                    // reuse (qp dead after step 3)
  float* h1    = qk;                    // reuse (qk dead after step 4)

  dim3 blk(128);

  // 1) pooled q
  pool_q<<<dim3(4 * 768 * 14), dim3(224), 0, stream>>>(x, q);

  // 2) qp = q @ wq.T + bq     (784x768x768)
  gemm_wmma_f32<<<dim3(13, 12, 1), blk, 0, stream>>>(
      q,    0, 0, 768, 1, 0, 0,
      w_in, 0, 0, 768, 1, 0, 0,
      qp,   0, 768, 1,
      b_in, 0,
      784, 768, 768, 1.0f, 0);

  // 3) qk_eff[h] = (qp_head @ wk_head) * scale   (batched over 12 heads, K=64)
  gemm_wmma_f32<<<dim3(13, 12, 12), blk, 0, stream>>>(
      qp,             64,    0, 768, 1,   0, 0,
      w_in + 589824,  49152, 0, 1,   768, 0, 0,     // wk[h*64+c][d]: rowStride(n=d)=1, kStride(c)=768
      qk,             768,   9216, 1,
      nullptr, 0,
      784, 768, 64, 0.125f, 0);                      // scale = 1/sqrt(64)

  // 4) scores[bn][l][h] = kv[l,:] . qk[bn][h,:]   (reads x directly, ATM addressing)
  gemm_wmma_f32<<<dim3(4, 1, 784), blk, 0, stream>>>(
      x,  0,    1, ROW3, 0,   0, 1,                  // A: rows l (stride 3*HW), k=c via atm_koff
      qk, 9216, 0, 768,  1,   0, 0,
      attn, 4096, 16, 1,
      nullptr, 0,
      256, 12, 768, 1.0f, 0);

  // 5) softmax over l
  softmax_attn<<<dim3(37), dim3(256), 0, stream>>>(attn);

  // 6) kvbar[bn][h][c] = sum_l attn[l,h] * kv[l,c]   (reads x again)
  gemm_wmma_f32<<<dim3(1, 12, 784), blk, 0, stream>>>(
      attn, 4096, 0, 1, 16,   0, 0,                  // A[m=h][k=l]
      x,    0,    1, 0, ROW3, 1, 0,                  // B[n=c][k=l]: rowOff=atm_koff(c), kStride=3*HW
      kvbar, 9216, 768, 1,
      nullptr, 0,
      12, 768, 256, 1.0f, 0);

  // 7) ctx[bn][h*64+cc] = kvbar[bn][h,:] @ wv_head.T + bv   (batched over heads)
  gemm_wmma_f32<<<dim3(13, 1, 12), blk, 0, stream>>>(
      kvbar,          768,   0, 9216, 1, 0, 0,
      w_in + 1179648, 49152, 0, 768,  1, 0, 0,
      ctx,            64,    768, 1,
      b_in + 1536,    64,
      784, 64, 768, 1.0f, 0);

  // 8) out = ctx @ w_out.T + b_out
  gemm_wmma_f32<<<dim3(13, 12, 1), blk, 0, stream>>>(
      ctx,   0, 0, 768, 1, 0, 0,
      w_out, 0, 0, 768, 1, 0, 0,
      outp,  0, 768, 1,
      b_out, 0,
      784, 768, 768, 1.0f, 0);

  // 9) h1 = silu(out @ w_fc1.T + b_fc1)
  gemm_wmma_f32<<<dim3(13, 48, 1), blk, 0, stream>>>(
      outp,  0, 0, 768, 1, 0, 0,
      w_fc1, 0, 0, 768, 1, 0, 0,
      h1,    0, 3072, 1,
      b_fc1, 0,
      784, 3072, 768, 1.0f, 1);

  // 10) y = h1 @ w_fc2.T + b_fc2
  gemm_wmma_f32<<<dim3(13, 12, 1), blk, 0, stream>>>(
      h1,    0, 0, 3072, 1, 0, 0,
      w_fc2, 0, 0, 3072, 1, 0, 0,
      y,     0, 768, 1,
      b_fc2, 0,
      784, 768, 3072, 1.0f, 0);
}